// GConv_40870908789472
// MI455X (gfx1250) — compile-verified
//
#include <hip/hip_runtime.h>
#include <hip/hip_bf16.h>
#include <stdint.h>

#define NN 100000
#define NE 1600000
#define F 128
#define BN_EPS 1e-5f

typedef __attribute__((ext_vector_type(2)))  float  v2f;
typedef __attribute__((ext_vector_type(8)))  float  v8f;

// ---------------- zero workspace (agg + stats) ----------------
__global__ void zero_kernel(float4* __restrict__ p, long n4) {
    long i = (long)blockIdx.x * blockDim.x + threadIdx.x;
    if (i < n4) p[i] = make_float4(0.f, 0.f, 0.f, 0.f);
}

// ---------------- SpMM: segmented accumulate over sorted rows ----------------
// One wave per 128-edge chunk. Lane l owns features [4l, 4l+4).
__global__ void spmm_kernel(const float* __restrict__ x,
                            const int*   __restrict__ row,
                            const int*   __restrict__ col,
                            const float* __restrict__ val,
                            float*       __restrict__ agg) {
    const int CHUNK = 128;
    int  wid  = (int)(((long)blockIdx.x * blockDim.x + threadIdx.x) >> 5);
    int  lane = threadIdx.x & 31;
    long start = (long)wid * CHUNK;
    if (start >= NE) return;
    long end = start + CHUNK;
    if (end > NE) end = NE;

    float4 acc = make_float4(0.f, 0.f, 0.f, 0.f);
    int cur = row[start];

    for (long base = start; base < end; base += 32) {
        int n = (int)(((end - base) < 32) ? (end - base) : 32);
        int r = 0, c = 0; float v = 0.f;
        if (lane < n) { r = row[base + lane]; c = col[base + lane]; v = val[base + lane]; }
        for (int j = 0; j < n; ++j) {
            int   rj = __shfl(r, j);
            int   cj = __shfl(c, j);
            float vj = __shfl(v, j);
            if (rj != cur) {
                float* p = agg + (long)cur * F + lane * 4;
                atomicAdd(p + 0, acc.x); atomicAdd(p + 1, acc.y);
                atomicAdd(p + 2, acc.z); atomicAdd(p + 3, acc.w);
                acc = make_float4(0.f, 0.f, 0.f, 0.f);
                cur = rj;
            }
            const float4 xv = *reinterpret_cast<const float4*>(x + (long)cj * F + lane * 4);
            acc.x += vj * xv.x; acc.y += vj * xv.y;
            acc.z += vj * xv.z; acc.w += vj * xv.w;
        }
    }
    float* p = agg + (long)cur * F + lane * 4;
    atomicAdd(p + 0, acc.x); atomicAdd(p + 1, acc.y);
    atomicAdd(p + 2, acc.z); atomicAdd(p + 3, acc.w);
}

// ---------------- GEMM (WMMA): h = agg @ W^T + b, plus per-feature partial stats ----
// Weights staged once per block into LDS, pre-swizzled into per-fragment lane order:
//   LDS element ((ks*8 + t)*32 + L)*2 + e  <-  W[n][k],
//   n = t*16 + (L&15), k = ks*4 + (L>>4)*2 + e.
// Each wave computes a 32(M) x 128(N) tile (2 M-tiles): every B fragment
// (one conflict-free ds_load_b64) feeds two v_wmma_f32_16x16x4_f32.
// __launch_bounds__(128,1): allow the full VGPR file (128 accumulator VGPRs live),
// preventing the scratch spills seen with the default occupancy cap.
__global__ void __launch_bounds__(128, 1)
gemm_kernel(const float* __restrict__ agg,
            const float* __restrict__ w,      // [128 out][128 in]
            const float* __restrict__ bias,
            float*       __restrict__ h,
            float*       __restrict__ sum,
            float*       __restrict__ sumsq,
            int mtiles32) {
    extern __shared__ float wlds[];                  // 16384 floats = 64 KB

    // ---- cooperative staging of W into fragment-ordered LDS ----
    for (int s = threadIdx.x; s < F * F; s += blockDim.x) {
        int n = s >> 7, k = s & 127;
        int t = n >> 4, nlo = n & 15;
        int ks = k >> 2, half = (k >> 1) & 1, e = k & 1;
        int L = half * 16 + nlo;
        wlds[((ks * 8 + t) * 32 + L) * 2 + e] = w[s];
    }
    __syncthreads();

    int wid  = (int)(((long)blockIdx.x * blockDim.x + threadIdx.x) >> 5);
    if (wid >= mtiles32) return;                     // whole-wave uniform exit after barrier
    int lane = threadIdx.x & 31;
    int nlo  = lane & 15;
    int half = lane >> 4;

    const float* arow0 = agg + ((long)wid * 32 + nlo) * F;        // M-tile 0
    const float* arow1 = arow0 + 16 * F;                          // M-tile 1

    v8f acc[2][8];
#pragma unroll
    for (int t = 0; t < 8; ++t) {
        float b = bias[t * 16 + nlo];
        v8f bv = {b, b, b, b, b, b, b, b};
        acc[0][t] = bv;
        acc[1][t] = bv;
    }

    for (int ks = 0; ks < 32; ++ks) {
        int k0 = ks * 4 + half * 2;
        v2f a0; a0.x = arow0[k0]; a0.y = arow0[k0 + 1];
        v2f a1; a1.x = arow1[k0]; a1.y = arow1[k0 + 1];
#pragma unroll
        for (int t = 0; t < 8; ++t) {
            v2f bf = *reinterpret_cast<const v2f*>(&wlds[((ks * 8 + t) * 32 + lane) * 2]);
            acc[0][t] = __builtin_amdgcn_wmma_f32_16x16x4_f32(
                false, a0, false, bf, (short)0, acc[0][t], false, false);
            acc[1][t] = __builtin_amdgcn_wmma_f32_16x16x4_f32(
                false, a1, false, bf, (short)0, acc[1][t], false, false);
        }
    }

    // D layout: lane -> N = nlo; VGPR i -> M = i + 8*half. Store + per-feature partials.
#pragma unroll
    for (int mt = 0; mt < 2; ++mt) {
        long mbase = (long)wid * 32 + mt * 16;
#pragma unroll
        for (int t = 0; t < 8; ++t) {
            int n = t * 16 + nlo;
            float s = 0.f, q = 0.f;
#pragma unroll
            for (int i = 0; i < 8; ++i) {
                float vv = acc[mt][t][i];
                long r = mbase + i + 8 * half;
                h[r * F + n] = vv;
                s += vv; q += vv * vv;
            }
            atomicAdd(&sum[n], s);
            atomicAdd(&sumsq[n], q);
        }
    }
}

// ---------------- BN stats -> scale/shift ----------------
__global__ void stats_kernel(const float* __restrict__ sum,
                             const float* __restrict__ sumsq,
                             const float* __restrict__ gamma,
                             const float* __restrict__ beta,
                             float* __restrict__ scale,
                             float* __restrict__ shift) {
    int t = threadIdx.x;
    if (t < F) {
        float inv_n = 1.0f / (float)NN;
        float mean = sum[t] * inv_n;
        float var  = sumsq[t] * inv_n - mean * mean;
        float sc   = gamma[t] * rsqrtf(var + BN_EPS);
        scale[t] = sc;
        shift[t] = beta[t] - mean * sc;
    }
}

// ---------------- apply BN in place on d_out ----------------
__global__ void apply_kernel(float* __restrict__ h,
                             const float* __restrict__ scale,
                             const float* __restrict__ shift,
                             long n4) {
    long i = (long)blockIdx.x * blockDim.x + threadIdx.x;
    if (i >= n4) return;
    float4 v = reinterpret_cast<float4*>(h)[i];
    int f = (int)((i * 4) & (F - 1));
    v.x = v.x * scale[f + 0] + shift[f + 0];
    v.y = v.y * scale[f + 1] + shift[f + 1];
    v.z = v.z * scale[f + 2] + shift[f + 2];
    v.w = v.w * scale[f + 3] + shift[f + 3];
    reinterpret_cast<float4*>(h)[i] = v;
}

extern "C" void kernel_launch(void* const* d_in, const int* in_sizes, int n_in,
                              void* d_out, int out_size, void* d_ws, size_t ws_size,
                              hipStream_t stream) {
    const float* x     = (const float*)d_in[0];
    const int*   arow  = (const int*)  d_in[1];
    const int*   acol  = (const int*)  d_in[2];
    const float* aval  = (const float*)d_in[3];
    const float* fc_w  = (const float*)d_in[4];
    const float* fc_b  = (const float*)d_in[5];
    const float* gamma = (const float*)d_in[6];
    const float* beta  = (const float*)d_in[7];
    float* out = (float*)d_out;

    float* ws    = (float*)d_ws;
    float* agg   = ws;                                // 12,800,000 floats
    float* sum   = ws + (long)NN * F;                 // 128
    float* sumsq = sum + F;                           // 128
    float* scale = sumsq + F;                         // 128
    float* shift = scale + F;                         // 128

    // 1) zero agg + sum + sumsq (12,800,256 floats = 3,200,064 float4)
    long n4z = ((long)NN * F + 2 * F) / 4;
    zero_kernel<<<(unsigned)((n4z + 255) / 256), 256, 0, stream>>>((float4*)ws, n4z);

    // 2) SpMM: 12500 waves (128 edges each), 8 waves/block
    int spmm_waves  = (NE + 127) / 128;
    int spmm_blocks = (spmm_waves + 7) / 8;
    spmm_kernel<<<spmm_blocks, 256, 0, stream>>>(x, arow, acol, aval, agg);

    // 3) GEMM + bias + partial BN stats: 3125 32-row tiles, 1 wave each, 4 waves/block
    int mtiles32 = NN / 32;                            // 3125 exact
    int gemm_blocks = (mtiles32 + 3) / 4;
    gemm_kernel<<<gemm_blocks, 128, 64 * 1024, stream>>>(agg, fc_w, fc_b, out, sum, sumsq, mtiles32);

    // 4) finalize BN scale/shift
    stats_kernel<<<1, 128, 0, stream>>>(sum, sumsq, gamma, beta, scale, shift);

    // 5) apply BN in place on d_out
    long n4 = (long)NN * F / 4;                        // 3,200,000
    apply_kernel<<<(unsigned)((n4 + 255) / 256), 256, 0, stream>>>(out, scale, shift, n4);
}